// generator_78108275245327
// MI455X (gfx1250) — compile-verified
//
#include <hip/hip_runtime.h>
#include <math.h>

// ---------------------------------------------------------------------------
// Shapes (fixed by the reference): L=200 slices, N=256 nodes, S=100 scan steps
// ---------------------------------------------------------------------------
#define NN       256
#define TSLICES  200
#define SSTEPS   100

typedef float v2f __attribute__((ext_vector_type(2)));
typedef float v8f __attribute__((ext_vector_type(8)));

// ---------------------------------------------------------------------------
// Kernel 1: f[s2*N + n] = exp(-dot(w[s2,n,:], e[s2,n,:]) * time[s2 & 1])
//
// Row-sum via fp32 WMMA: D = A(16x4) * ones(4x16) + C  accumulated over K=256
// (64 chained v_wmma_f32_16x16x4_f32). Every column of D holds the row sum,
// exact in fp32. One wave = one 16-row block of one slice. EXEC is all-ones
// for the whole WMMA loop (grid sized exactly, no early exits).
// ---------------------------------------------------------------------------
__global__ __launch_bounds__(256) void fdot_wmma_kernel(
    const float* __restrict__ ev,    // [T+1, N, N] (only first T slices used)
    const float* __restrict__ w,     // [T, N, 1, N] == [T, N, N]
    const float* __restrict__ time2, // [2]
    float* __restrict__ fbuf)        // [T, N]
{
    const int wave = (int)((blockIdx.x * blockDim.x + threadIdx.x) >> 5);
    const int lane = (int)(threadIdx.x & 31);
    const int s2 = wave >> 4;        // slice index 0..199
    const int rb = wave & 15;        // 16-row block 0..15
    const int r0 = rb * 16;

    // A-matrix 16x4 fp32 layout (ISA 7.12.2):
    //   lanes 0-15 : M=lane,    v0=K+0, v1=K+1
    //   lanes 16-31: M=lane-16, v0=K+2, v1=K+3
    const int row    = r0 + (lane & 15);
    const int colsel = (lane >> 4) * 2;
    const size_t base = ((size_t)s2 * NN + row) * NN;

    v2f bones = {1.0f, 1.0f};        // all-ones B (4x16) -> row sums in D
    v8f acc = {};

    #pragma unroll 4
    for (int k = 0; k < NN; k += 4) {
        const float* wp = w  + base + k + colsel;
        const float* ep = ev + base + k + colsel;
        v2f av;
        av.x = wp[0] * ep[0];
        av.y = wp[1] * ep[1];
        acc = __builtin_amdgcn_wmma_f32_16x16x4_f32(
            /*neg_a=*/false, av, /*neg_b=*/false, bones,
            /*c_mod=*/(short)0, acc, /*reuse_a=*/false, /*reuse_b=*/false);
    }

    // D layout: VGPR v, lanes 0-15 -> row v; lanes 16-31 -> row v+8.
    // All columns identical, so lane 0 emits rows r0..r0+7, lane 16 emits
    // rows r0+8..r0+15.
    const float t = time2[s2 & 1];
    if (lane == 0 || lane == 16) {
        const int rbase = r0 + ((lane >> 4) << 3);
        #pragma unroll
        for (int i = 0; i < 8; ++i) {
            fbuf[s2 * NN + rbase + i] = __expf(-acc[i] * t) ;
        }
    }
}

// ---------------------------------------------------------------------------
// Kernel 2: the 100-step scan. Block = node n (f, liner_w, liner_b uniform
// per block), thread = column m. State u[3] lives in registers across the
// whole chain. `a` is streamed once -> non-temporal loads; next step's slice
// is prefetched (global_prefetch_b8) to hide latency at 2048-wave occupancy.
// ---------------------------------------------------------------------------
__global__ __launch_bounds__(256) void scan_kernel(
    const float* __restrict__ u_begin, // [N, N, 3]
    const float* __restrict__ a,       // [T, N, N, 3]
    const float* __restrict__ fbuf,    // [T, N]
    const float* __restrict__ liner_w, // [N, 3]
    const float* __restrict__ liner_b, // [N]
    float* __restrict__ out)           // [S, N, N]
{
    const int n = (int)blockIdx.x;     // 0..255
    const int m = (int)threadIdx.x;    // 0..255

    const float lw0 = liner_w[n * 3 + 0];
    const float lw1 = liner_w[n * 3 + 1];
    const float lw2 = liner_w[n * 3 + 2];
    const float bb  = liner_b[n];

    const size_t uo = ((size_t)n * NN + m) * 3;
    float u0 = u_begin[uo + 0];
    float u1 = u_begin[uo + 1];
    float u2 = u_begin[uo + 2];

    const size_t slice = (size_t)NN * NN * 3;          // elems per a-slice
    size_t a0o = (((size_t)n) * NN + m) * 3;           // a[0, n, m, 0]

    for (int s = 0; s < SSTEPS; ++s) {
        const int s2 = 2 * s;
        const float f0 = fbuf[s2 * NN + n];            // uniform per block
        const float f1 = fbuf[(s2 + 1) * NN + n];

        const float* pa0 = a + a0o;
        const float* pa1 = pa0 + slice;

        if (s + 1 < SSTEPS) {
            __builtin_prefetch(pa0 + 2 * slice, 0, 1); // -> global_prefetch_b8
            __builtin_prefetch(pa1 + 2 * slice, 0, 1);
        }

        // streamed-once data: non-temporal loads (TH=NT), keep L2 for f/out
        const float a00 = __builtin_nontemporal_load(pa0 + 0);
        const float a01 = __builtin_nontemporal_load(pa0 + 1);
        const float a02 = __builtin_nontemporal_load(pa0 + 2);
        const float a10 = __builtin_nontemporal_load(pa1 + 0);
        const float a11 = __builtin_nontemporal_load(pa1 + 1);
        const float a12 = __builtin_nontemporal_load(pa1 + 2);

        const float l0 = fmaf(a10, f1, fmaf(a00, f0, u0));
        const float l1 = fmaf(a11, f1, fmaf(a01, f0, u1));
        const float l2 = fmaf(a12, f1, fmaf(a02, f0, u2));

        // softmax over the 3 channels (max-subtracted, as jax.nn.softmax)
        const float mx = fmaxf(l0, fmaxf(l1, l2));
        const float e0 = __expf(l0 - mx);
        const float e1 = __expf(l1 - mx);
        const float e2 = __expf(l2 - mx);
        const float inv = 1.0f / (e0 + e1 + e2);
        u0 = e0 * inv;
        u1 = e1 * inv;
        u2 = e2 * inv;

        const float o = fmaf(u0, lw0, fmaf(u1, lw1, fmaf(u2, lw2, bb)));
        __builtin_nontemporal_store(o, out + (size_t)s * NN * NN + n * NN + m);

        a0o += 2 * slice;
    }
}

// ---------------------------------------------------------------------------
// Launch: inputs in setup_inputs() order:
//   0 time[2], 1 event_list[201,256,256], 2 u_begin[256,256,3],
//   3 a[200,256,256,3], 4 w[200,256,1,256], 5 liner_w[256,3],
//   6 liner_b[256], 7 batch_size (==2, unused)
// Output: [100, 256, 256] float32.  Workspace: fbuf = 200*256 floats.
// ---------------------------------------------------------------------------
extern "C" void kernel_launch(void* const* d_in, const int* in_sizes, int n_in,
                              void* d_out, int out_size, void* d_ws, size_t ws_size,
                              hipStream_t stream) {
    (void)in_sizes; (void)n_in; (void)out_size; (void)ws_size;

    const float* time2   = (const float*)d_in[0];
    const float* ev      = (const float*)d_in[1];
    const float* u_begin = (const float*)d_in[2];
    const float* a       = (const float*)d_in[3];
    const float* w       = (const float*)d_in[4];
    const float* lw      = (const float*)d_in[5];
    const float* lb      = (const float*)d_in[6];
    float* out  = (float*)d_out;
    float* fbuf = (float*)d_ws;            // 200*256*4 = 200 KB scratch

    // 200 slices * 16 row-blocks = 3200 waves = 102400 threads (exact)
    fdot_wmma_kernel<<<400, 256, 0, stream>>>(ev, w, time2, fbuf);
    // one block per node n, one thread per column m
    scan_kernel<<<NN, NN, 0, stream>>>(u_begin, a, fbuf, lw, lb, out);
}